// Model_42296837931423
// MI455X (gfx1250) — compile-verified
//
#include <hip/hip_runtime.h>
#include <math.h>

typedef __attribute__((ext_vector_type(2))) float v2f;
typedef __attribute__((ext_vector_type(8))) float v8f;

#define NSTATE 2048
#define NB     16
#define NVERT  6
#define TSTEPS 128
#define NBLK   128          // grid: one 16-column output tile per block
#define NTHR   128          // 4 waves per block

// State layout (K-pair interleaved so the A operand is one b64 per lane):
//   wq[(i>>1)*32 + b*2 + (i&1)] = p[b,i] * mask
#define WIDX(i, b) ((((i) >> 1) << 5) + ((b) << 1) + ((i) & 1))

// ---------------------------------------------------------------------------
// Device-scope grid barrier: monotonic counter, release on arrive, acquire on
// observe.  128 tiny blocks are trivially co-resident on MI455X.
// ---------------------------------------------------------------------------
__device__ __forceinline__ void gsync(unsigned* bar, unsigned target) {
    __syncthreads();
    if (threadIdx.x == 0) {
        __hip_atomic_fetch_add(bar, 1u, __ATOMIC_RELEASE, __HIP_MEMORY_SCOPE_AGENT);
        while (__hip_atomic_load(bar, __ATOMIC_ACQUIRE, __HIP_MEMORY_SCOPE_AGENT) < target)
            __builtin_amdgcn_s_sleep(1);
    }
    __syncthreads();
}

// ---------------------------------------------------------------------------
// One fused persistent kernel:
//  Phase A: row softmax of both logit matrices  -> Pno, Pyes (L2-resident)
//  Phase B: initial masked interleaved state (p = delta_0)
//  Phase C: 128 Markov steps, each = two (16x2048)x(2048x2048) fp32-WMMA
//           GEMM slices per block, grid barrier between steps
//  Phase D: blocks 0..15 compute log p[b,N-2], log p[b,N-1], entropy
// ---------------------------------------------------------------------------
__global__ void markov_fused_kernel(const int* __restrict__ graphs, const int* __restrict__ Q,
                                    const float* __restrict__ Lno, const float* __restrict__ Lyes,
                                    float* __restrict__ Pno, float* __restrict__ Pyes,
                                    float* __restrict__ wno0, float* __restrict__ wyes0,
                                    float* __restrict__ wno1, float* __restrict__ wyes1,
                                    unsigned* __restrict__ bar,
                                    float* __restrict__ out) {
    __shared__ float sbuf[4 * 256];   // step reduction (4KB); reused by softmax/finalize
    const int tid = threadIdx.x;

    // ---------------- Phase A: softmax rows (32 rows per block) --------------
    for (int r = blockIdx.x; r < 2 * NSTATE; r += NBLK) {
        const float* Lr = (r < NSTATE) ? (Lno + (size_t)r * NSTATE)
                                       : (Lyes + (size_t)(r - NSTATE) * NSTATE);
        float* Pr = (r < NSTATE) ? (Pno + (size_t)r * NSTATE)
                                 : (Pyes + (size_t)(r - NSTATE) * NSTATE);
        float m = -INFINITY;
        for (int j = tid; j < NSTATE; j += NTHR) m = fmaxf(m, Lr[j]);
        sbuf[tid] = m;
        __syncthreads();
        for (int s = NTHR / 2; s > 0; s >>= 1) {
            if (tid < s) sbuf[tid] = fmaxf(sbuf[tid], sbuf[tid + s]);
            __syncthreads();
        }
        m = sbuf[0];
        __syncthreads();
        float sum = 0.f;
        for (int j = tid; j < NSTATE; j += NTHR) sum += __expf(Lr[j] - m);
        sbuf[tid] = sum;
        __syncthreads();
        for (int s = NTHR / 2; s > 0; s >>= 1) {
            if (tid < s) sbuf[tid] += sbuf[tid + s];
            __syncthreads();
        }
        const float inv = 1.0f / sbuf[0];
        __syncthreads();
        for (int j = tid; j < NSTATE; j += NTHR) Pr[j] = __expf(Lr[j] - m) * inv;
    }

    // ---------------- Phase B: initial state (p = delta_0, masked) -----------
    for (int idx = blockIdx.x * NTHR + tid; idx < NSTATE * NB; idx += NBLK * NTHR) {
        // idx = m*32 + b*2 + v  ->  i = 2m+v
        const int b = (idx >> 1) & 15;
        const int i = ((idx >> 5) << 1) | (idx & 1);
        float vy = 0.f, vn = 0.f;
        if (i == 0) {
            const int q0 = Q[0], q1 = Q[1];
            const float a = (float)graphs[b * (NVERT * NVERT) + q0 * NVERT + q1];
            vy = a;
            vn = 1.0f - a;
        }
        wyes0[idx] = vy;
        wno0[idx]  = vn;
    }

    unsigned tgt = NBLK;
    gsync(bar, tgt);

    // ---------------- Phase C setup (all loop-invariant) ---------------------
    const int lane = tid & 31;
    const int wave = tid >> 5;        // 0..3, K-slice of 512 each
    const int lcol = lane & 15;
    const int half = lane >> 4;
    const int j0   = blockIdx.x * 16;
    const int kbase = wave * (NSTATE / 4) + 2 * half;

    const float* bno0  = Pno  + (size_t)kbase * NSTATE + j0 + lcol;
    const float* byes0 = Pyes + (size_t)kbase * NSTATE + j0 + lcol;
    // A operand: one b64 per quad at wq + aoff0 + q*64
    const int aoff0 = (wave * 256 + half) * 32 + lcol * 2;

    // Epilogue constants: this thread's two C/D elements (e, e+128).
    // C/D layout: (VGPR r, lane l): l<16 -> M=r,N=l ; l>=16 -> M=r+8,N=l-16.
    const int e0 = tid;
    const int l0 = e0 & 31;
    const int M0 = (e0 >> 5) + ((l0 >> 4) << 3);
    const int M1 = M0 + 4;                       // e1 = e0 + 128 -> r += 4
    const int col = j0 + (l0 & 15);              // same column for both
    const int off0 = WIDX(col, M0);
    const int off1 = WIDX(col, M1);
    const int qq0 = Q[col * 2 + 0], qq1 = Q[col * 2 + 1];
    const float a0  = (float)graphs[M0 * (NVERT * NVERT) + qq0 * NVERT + qq1];
    const float a1  = (float)graphs[M1 * (NVERT * NVERT) + qq0 * NVERT + qq1];
    const float na0 = 1.0f - a0, na1 = 1.0f - a1;

    const float* wns = wno0;  const float* wys = wyes0;
    float*       wnd = wno1;  float*       wyd = wyes1;

    // ---------------- Phase C: 128 Markov steps ------------------------------
    for (int t = 0; t < TSTEPS; ++t) {
        const float* bno  = bno0;
        const float* byes = byes0;
        const float* ano  = wns + aoff0;
        const float* ayes = wys + aoff0;

        v8f accN = {};
        v8f accY = {};
#pragma unroll 8
        for (int q = 0; q < NSTATE / 16; ++q) {   // 128 K-quads of 4
            v2f aN = *(const v2f*)ano;            // one b64: K pair (v=0,1)
            v2f aY = *(const v2f*)ayes;
            v2f bN, bY;
            bN.x = bno[0];   bN.y = bno[NSTATE];
            bY.x = byes[0];  bY.y = byes[NSTATE];
            accN = __builtin_amdgcn_wmma_f32_16x16x4_f32(false, aN, false, bN,
                                                         (short)0, accN, false, false);
            accY = __builtin_amdgcn_wmma_f32_16x16x4_f32(false, aY, false, bY,
                                                         (short)0, accY, false, false);
            ano += 64;          ayes += 64;          // 2 K-pairs * 32
            bno += 4 * NSTATE;  byes += 4 * NSTATE;
        }

        v8f acc = accN + accY;
#pragma unroll
        for (int r = 0; r < 8; ++r) sbuf[wave * 256 + r * 32 + lane] = acc[r];
        __syncthreads();

        // Sum 4 partial tiles and write masked state for next step.
        const float s0 = sbuf[e0] + sbuf[256 + e0] + sbuf[512 + e0] + sbuf[768 + e0];
        const int   e1 = e0 + 128;
        const float s1 = sbuf[e1] + sbuf[256 + e1] + sbuf[512 + e1] + sbuf[768 + e1];
        wyd[off0] = s0 * a0;  wnd[off0] = s0 * na0;
        wyd[off1] = s1 * a1;  wnd[off1] = s1 * na1;

        tgt += NBLK;
        gsync(bar, tgt);

        // ping-pong
        const float* tn = wns; wns = wnd; wnd = (float*)tn;
        const float* ty = wys; wys = wyd; wyd = (float*)ty;
    }

    // ---------------- Phase D: outputs (final state is in buffer 0) ----------
    if (blockIdx.x < NB) {
        const int b = blockIdx.x;
        float H = 0.f;
        for (int j = tid; j < NSTATE; j += NTHR) {
            const float p = wno0[WIDX(j, b)] + wyes0[WIDX(j, b)];
            const float lp = __logf(p);
            if (p > 0.f) H -= p * lp;
            if (j == NSTATE - 2) out[b] = lp;
            if (j == NSTATE - 1) out[NB + b] = lp;
        }
        sbuf[tid] = H;
        __syncthreads();
        for (int s = NTHR / 2; s > 0; s >>= 1) {
            if (tid < s) sbuf[tid] += sbuf[tid + s];
            __syncthreads();
        }
        if (tid == 0) out[2 * NB + b] = sbuf[0];
    }
}

// ---------------------------------------------------------------------------
extern "C" void kernel_launch(void* const* d_in, const int* in_sizes, int n_in,
                              void* d_out, int out_size, void* d_ws, size_t ws_size,
                              hipStream_t stream) {
    (void)in_sizes; (void)n_in; (void)out_size; (void)ws_size;
    const int*   graphs = (const int*)d_in[0];
    const int*   Q      = (const int*)d_in[1];
    const float* Lno    = (const float*)d_in[2];
    const float* Lyes   = (const float*)d_in[3];
    float* out = (float*)d_out;

    char* ws = (char*)d_ws;
    size_t off = 0;
    auto wsalloc = [&](size_t bytes) -> void* {
        void* p = ws + off;
        off = (off + bytes + 255) & ~(size_t)255;
        return p;
    };
    unsigned* bar = (unsigned*)wsalloc(256);
    float* Pno   = (float*)wsalloc((size_t)NSTATE * NSTATE * sizeof(float)); // 16.8 MB
    float* Pyes  = (float*)wsalloc((size_t)NSTATE * NSTATE * sizeof(float)); // 16.8 MB
    float* wno0  = (float*)wsalloc((size_t)NSTATE * NB * sizeof(float));
    float* wyes0 = (float*)wsalloc((size_t)NSTATE * NB * sizeof(float));
    float* wno1  = (float*)wsalloc((size_t)NSTATE * NB * sizeof(float));
    float* wyes1 = (float*)wsalloc((size_t)NSTATE * NB * sizeof(float));

    hipMemsetAsync(bar, 0, 256, stream);
    markov_fused_kernel<<<NBLK, NTHR, 0, stream>>>(graphs, Q, Lno, Lyes,
                                                   Pno, Pyes,
                                                   wno0, wyes0, wno1, wyes1,
                                                   bar, out);
}